// ModConv2d_19971597926864
// MI455X (gfx1250) — compile-verified
//
#include <hip/hip_runtime.h>
#include <math.h>

typedef __attribute__((ext_vector_type(2))) float v2f;
typedef __attribute__((ext_vector_type(8))) float v8f;
typedef __attribute__((ext_vector_type(4))) unsigned v4u;
typedef __attribute__((ext_vector_type(8))) unsigned v8u;

#define N_    8
#define CIN   64
#define COUT  64
#define H_    256
#define W_    256
#define AUX   128
#define HID   256
#define KTAPS 9

// workspace layout (float offsets)
#define WS_HM    0
#define WS_HB    (WS_HM + N_*HID)          // 2048
#define WS_BIAS  (WS_HB + N_*HID)          // 4096
#define WS_APACK (WS_BIAS + N_*COUT)       // 4608
// Apack: N_ * KTAPS * 32 * 128 floats = 294912 floats (~1.15 MB)

#define TILE_W   64
#define XS_COLS  66   // w0-1 .. w0+64
#define XP_PITCH 66   // cin dim 64 padded to 66 -> conflict-free, 8B-aligned b64

// ---------- prep 1: hidden layers of both MLPs (PReLU) ----------
__global__ void mlp_hidden_kernel(const float* __restrict__ y,
                                  const float* __restrict__ fc_w1,
                                  const float* __restrict__ fc_b1,
                                  const float* __restrict__ fc_prelu,
                                  const float* __restrict__ bias_w1,
                                  const float* __restrict__ bias_b1,
                                  const float* __restrict__ bias_prelu,
                                  float* __restrict__ ws) {
  int t = blockIdx.x * blockDim.x + threadIdx.x;   // 0..4095
  int sel = t >> 11;                               // 0: modulation MLP, 1: bias MLP
  int r = t & 2047;
  int n = r >> 8;
  int j = r & 255;
  const float* w1 = sel ? bias_w1 : fc_w1;
  const float* b1 = sel ? bias_b1 : fc_b1;
  float alpha = sel ? bias_prelu[0] : fc_prelu[0];
  const float* yr = y + n * AUX;
  const float* wr = w1 + j * AUX;
  float acc = b1[j];
  #pragma unroll 4
  for (int k = 0; k < AUX; ++k) acc += yr[k] * wr[k];
  float h = acc >= 0.f ? acc : alpha * acc;
  ws[(sel ? WS_HB : WS_HM) + n * HID + j] = h;
}

// ---------- prep 2: modulation (sigmoid) + WMMA-friendly weight packing ----------
__global__ void mod_pack_kernel(const float* __restrict__ weight,
                                const float* __restrict__ fc_w2,
                                const float* __restrict__ fc_b2,
                                float* __restrict__ ws) {
  int t = blockIdx.x * blockDim.x + threadIdx.x;   // 0..32767
  int n = t >> 12;
  int r = t & 4095;                                // cout*64 + cin
  int cout = r >> 6;
  int cin  = r & 63;
  const float* hm = ws + WS_HM + n * HID;
  const float* wr = fc_w2 + (size_t)r * HID;
  float acc = fc_b2[r];
  #pragma unroll 4
  for (int k = 0; k < HID; ++k) acc += hm[k] * wr[k];
  float s = 1.f / (1.f + __expf(-acc));
  float* ap = ws + WS_APACK;
  const float* wk = weight + (size_t)r * KTAPS;
  #pragma unroll
  for (int tap = 0; tap < KTAPS; ++tap) {
    int idx = (((n * KTAPS + tap) * 32 + (cin >> 1)) * COUT + cout) * 2 + (cin & 1);
    ap[idx] = s * wk[tap];
  }
}

// ---------- prep 3: per-sample learned bias ----------
__global__ void bias_kernel(const float* __restrict__ bias_w2,
                            const float* __restrict__ bias_b2,
                            float* __restrict__ ws) {
  int t = blockIdx.x * blockDim.x + threadIdx.x;   // 0..511
  int n = t >> 6;
  int c = t & 63;
  const float* hb = ws + WS_HB + n * HID;
  const float* wr = bias_w2 + (size_t)c * HID;
  float acc = bias_b2[c];
  #pragma unroll 4
  for (int k = 0; k < HID; ++k) acc += hb[k] * wr[k];
  ws[WS_BIAS + t] = acc;
}

// ---------- main conv: implicit GEMM, V_WMMA_F32_16X16X4_F32 + TDM staging ----------
__global__ __launch_bounds__(256)
void modconv_wmma_kernel(const float* __restrict__ x,
                         const float* __restrict__ ws,
                         float* __restrict__ out) {
  __shared__ float rawb[CIN * 3 * XS_COLS];           // TDM landing: [cin][row][col]
  __shared__ float xp[3 * XS_COLS * XP_PITCH];        // compute: [row][col][cin+pad]
  __shared__ float bs[COUT];

  int w0 = blockIdx.x * TILE_W;
  int h  = blockIdx.y;
  int n  = blockIdx.z;
  int tid = threadIdx.x;

  if (tid < COUT) bs[tid] = ws[WS_BIAS + n * COUT + tid];

  bool interior = (blockIdx.x > 0) && (blockIdx.y > 0);
  if (interior) {
    if (tid < 32) {                                   // wave 0 issues the DMA
      int hstart = h - 1;                             // >= 0
      int wstart = w0 - 1;                            // >= 63
      const float* gptr = x + (size_t)n * CIN * H_ * W_
                            + (size_t)hstart * W_ + wstart;
      unsigned long long ga = (unsigned long long)(uintptr_t)gptr;
      unsigned ldsa = (unsigned)(uintptr_t)(void*)rawb;
      unsigned td0 = (unsigned)(W_ - wstart);         // OOB right -> zeros
      unsigned td1 = (unsigned)(H_ - hstart);         // OOB bottom -> zeros
      // D# group0: count=1, lds_addr, global_addr, type=2
      v4u g0 = { 1u, ldsa, (unsigned)ga, (unsigned)(ga >> 32) | 0x80000000u };
      // D# group1: data_size=4B; dims/strides; tile 66 x 3 x 64
      v8u g1 = { 0x20000u,
                 (td0 & 0xFFFFu) << 16,
                 (td0 >> 16) | ((td1 & 0xFFFFu) << 16),
                 (td1 >> 16) | ((unsigned)XS_COLS << 16),
                 3u | (64u << 16),
                 (unsigned)W_,                         // dim0 stride (row)
                 0u,
                 (unsigned)((H_ * W_) >> 16) };        // dim1 stride (cin) = 65536
      v4u g2 = { 64u, 0u, 0u, 0u };                    // tensor_dim2 = CIN
      v4u g3 = { 0u, 0u, 0u, 0u };
      asm volatile("tensor_load_to_lds %0, %1, %2, %3"
                   :: "s"(g0), "s"(g1), "s"(g2), "s"(g3)
                   : "memory");
      __builtin_amdgcn_s_wait_tensorcnt(0);
    }
    __syncthreads();
    // transpose raw [cin][row][col] -> xp [row][col][cin]
    for (int i = tid; i < CIN * 3 * XS_COLS; i += 256) {
      int cin = i / (3 * XS_COLS);
      int rc  = i - cin * (3 * XS_COLS);              // row*66 + col
      xp[rc * XP_PITCH + cin] = rawb[i];
    }
    __syncthreads();
  } else {
    // border: predicated global fill straight into compute layout
    const float* xn = x + (size_t)n * CIN * H_ * W_;
    for (int i = tid; i < CIN * 3 * XS_COLS; i += 256) {
      int cin = i / (3 * XS_COLS);
      int rc  = i - cin * (3 * XS_COLS);
      int dh  = rc / XS_COLS;
      int col = rc - dh * XS_COLS;
      int gh = h + dh - 1;
      int gw = w0 + col - 1;
      float v = 0.f;
      if ((unsigned)gh < (unsigned)H_ && (unsigned)gw < (unsigned)W_)
        v = xn[((size_t)cin * H_ + gh) * W_ + gw];
      xp[rc * XP_PITCH + cin] = v;
    }
    __syncthreads();
  }

  int lane  = tid & 31;
  int wave  = tid >> 5;
  int waveM = wave >> 1;           // 0..3 : 16-cout tile
  int waveP = wave & 1;            // 0..1 : 32-pixel group
  int m16   = lane & 15;
  int hi    = lane >> 4;
  int koff  = hi * 2;
  int coutBase = waveM * 16;
  int m     = coutBase + m16;
  int p0    = waveP * 32 + m16;

  const float* ap = ws + WS_APACK + (size_t)n * KTAPS * 32 * 128;
  v8f c0 = {}, c1 = {};
  #pragma unroll
  for (int tap = 0; tap < KTAPS; ++tap) {
    int dh = tap / 3;
    int dw = tap - 3 * dh;
    const float* b0Base = &xp[(dh * XS_COLS + p0 + dw) * XP_PITCH + koff];
    const float* b1Base = b0Base + 16 * XP_PITCH;     // p1 = p0 + 16
    const float* apBase = ap + (size_t)tap * 4096 + m * 2 + hi * 128;
    #pragma unroll
    for (int cinb = 0; cinb < CIN; cinb += 4) {
      v2f a  = *(const v2f*)(apBase + (size_t)(cinb >> 1) * 128);  // coalesced 8B
      v2f b0 = *(const v2f*)(b0Base + cinb);                       // LDS b64
      v2f b1 = *(const v2f*)(b1Base + cinb);
      c0 = __builtin_amdgcn_wmma_f32_16x16x4_f32(
               false, a, false, b0, (short)0, c0, false, false);
      c1 = __builtin_amdgcn_wmma_f32_16x16x4_f32(
               false, a, false, b1, (short)0, c1, false, false);
    }
  }

  // D layout: VGPR j holds M = j + 8*hi, N = lane&15
  float* on = out + (size_t)n * COUT * H_ * W_;
  int wA = w0 + p0;
  int wB = wA + 16;
  #pragma unroll
  for (int j = 0; j < 8; ++j) {
    int cout = coutBase + hi * 8 + j;
    float b = bs[cout];
    size_t rowbase = ((size_t)cout * H_ + h) * W_;
    on[rowbase + wA] = c0[j] + b;
    on[rowbase + wB] = c1[j] + b;
  }
}

extern "C" void kernel_launch(void* const* d_in, const int* in_sizes, int n_in,
                              void* d_out, int out_size, void* d_ws, size_t ws_size,
                              hipStream_t stream) {
  const float* x          = (const float*)d_in[0];
  const float* y          = (const float*)d_in[1];
  const float* weight     = (const float*)d_in[2];
  const float* fc_w1      = (const float*)d_in[3];
  const float* fc_b1      = (const float*)d_in[4];
  const float* fc_prelu   = (const float*)d_in[5];
  const float* fc_w2      = (const float*)d_in[6];
  const float* fc_b2      = (const float*)d_in[7];
  const float* bias_w1    = (const float*)d_in[8];
  const float* bias_b1    = (const float*)d_in[9];
  const float* bias_prelu = (const float*)d_in[10];
  const float* bias_w2    = (const float*)d_in[11];
  const float* bias_b2    = (const float*)d_in[12];
  float* out = (float*)d_out;
  float* ws  = (float*)d_ws;

  mlp_hidden_kernel<<<16, 256, 0, stream>>>(y, fc_w1, fc_b1, fc_prelu,
                                            bias_w1, bias_b1, bias_prelu, ws);
  mod_pack_kernel<<<128, 256, 0, stream>>>(weight, fc_w2, fc_b2, ws);
  bias_kernel<<<2, 256, 0, stream>>>(bias_w2, bias_b2, ws);

  dim3 grid(W_ / TILE_W, H_, N_);
  modconv_wmma_kernel<<<grid, 256, 0, stream>>>(x, ws, out);
}